// Bert_Proj_CRF_2267742732599
// MI455X (gfx1250) — compile-verified
//
#include <hip/hip_runtime.h>
#include <hip/hip_bf16.h>

#define VOCAB 21128
#define DD 768
#define TT 4
#define NC 10
#define BB 64
#define SS 512

// Padded LDS row stride: 772 mod 64 = 4 -> consecutive rows land 4 banks
// apart (no multi-lane bank conflict); 772*4B is 16B aligned so b64 LDS
// reads stay legal.
#define WSTRIDE 772
#define WROWS 5   // rows 0..3 = classes, row 4 = zeros (for lanes n>=4)

typedef float v2f __attribute__((ext_vector_type(2)));
typedef float v8f __attribute__((ext_vector_type(8)));

// ds_swizzle group-of-32 encodings: offset = xor<<10 | or<<5 | and
#define SWZ_X1 0x041f  // lane ^= 1
#define SWZ_X2 0x081f  // lane ^= 2
#define SWZF(v, imm) \
    __int_as_float(__builtin_amdgcn_ds_swizzle(__float_as_int(v), (imm)))

// ---------------------------------------------------------------------------
// Kernel 0: precompute combined (transposed) weights per domain:
//   WcT[c][t][d] = shared_W[d][t] + domain_A[c][d][t]
//   bc[c][t]     = shared_b[t]    + domain_b[c][t]
// ---------------------------------------------------------------------------
__global__ void prep_kernel(const float* __restrict__ shared_W,
                            const float* __restrict__ shared_b,
                            const float* __restrict__ domain_A,
                            const float* __restrict__ domain_b,
                            float* __restrict__ WcT,
                            float* __restrict__ bc) {
    int i = blockIdx.x * blockDim.x + threadIdx.x;
    if (i < NC * TT * DD) {
        int d = i % DD;
        int t = (i / DD) % TT;
        int c = i / (DD * TT);
        WcT[i] = shared_W[d * TT + t] + domain_A[(c * DD + d) * TT + t];
    }
    if (i < NC * TT) {
        bc[i] = shared_b[i % TT] + domain_b[i];
    }
}

// ---------------------------------------------------------------------------
// Kernel 1: fused embedding-gather + projection (WMMA f32 16x16x4) +
//           log_softmax over T=4.  One wave handles 16 tokens; a 256-thread
//           workgroup (8 waves) handles 128 tokens, all in the same batch b,
//           sharing a 15KB combined-weight panel (4 class rows + 1 zero row)
//           in LDS.
// ---------------------------------------------------------------------------
__global__ __launch_bounds__(256) void proj_kernel(
        const int*   __restrict__ words,
        const int*   __restrict__ corpus,
        const float* __restrict__ embed,
        const float* __restrict__ WcT,
        const float* __restrict__ bc,
        float*       __restrict__ logits) {
    __shared__ float sW[WROWS * WSTRIDE];   // [row][d], row 4 = zeros
    __shared__ float sb[TT];

    const int wg = blockIdx.x;            // 256 workgroups, 128 tokens each
    const int b  = (wg * 128) / SS;       // batch index (all tokens same b)
    const int c  = corpus[b];

    // Load weight panel; zero row 4 and all pad columns.
    for (int i = threadIdx.x; i < WROWS * WSTRIDE; i += 256) {
        int r = i / WSTRIDE;
        int d = i - r * WSTRIDE;
        sW[i] = (r < TT && d < DD) ? WcT[(c * TT + r) * DD + d] : 0.0f;
    }
    if (threadIdx.x < TT) sb[threadIdx.x] = bc[c * TT + threadIdx.x];
    __syncthreads();

    const int wave = threadIdx.x >> 5;
    const int lane = threadIdx.x & 31;
    const int base = wg * 128 + wave * 16;     // first token of this wave's tile
    const int m    = lane & 15;                // row (token) / col (class) index
    const int koff = (lane & 16) ? 2 : 0;      // K sub-offset per ISA A/B layout

    const int    token = base + m;
    const int    word  = words[token];
    const float* row   = embed + (long)word * DD;

    const bool   valid = (m < TT);
    const int    widx  = valid ? m : TT;       // lanes n>=4 stream the zero row
    const float* wrow  = &sW[widx * WSTRIDE];

    v8f acc = {};
#pragma unroll 8
    for (int k0 = 0; k0 < DD; k0 += 4) {
        v2f a  = *(const v2f*)(row  + k0 + koff);   // 8B aligned (even index)
        v2f bf = *(const v2f*)(wrow + k0 + koff);   // zero row for n>=4
        acc = __builtin_amdgcn_wmma_f32_16x16x4_f32(
                  /*neg_a=*/false, a, /*neg_b=*/false, bf,
                  /*c_mod=*/(short)0, acc, /*reuse_a=*/false, /*reuse_b=*/false);
    }

    // acc VGPR r holds out[M = r + (lane>=16 ? 8 : 0)][N = lane&15].
    // log_softmax across N=0..3, i.e. across lanes {0..3} / {16..19}:
    // xor-1 / xor-2 ds_swizzle butterflies stay inside each 4-lane group.
    const float bias = sb[valid ? m : 0];
#pragma unroll
    for (int r = 0; r < 8; ++r) {
        float v  = acc[r] + bias;
        float mx = fmaxf(v, SWZF(v, SWZ_X1));
        mx       = fmaxf(mx, SWZF(mx, SWZ_X2));
        float e  = __expf(v - mx);
        float sm = e + SWZF(e, SWZ_X1);
        sm       = sm + SWZF(sm, SWZ_X2);
        float lsm = v - mx - __logf(sm);
        if (valid) {
            int tok = base + r + ((lane & 16) ? 8 : 0);
            logits[tok * TT + m] = lsm;
        }
    }
}

// ---------------------------------------------------------------------------
// Kernel 2: CRF forward scan + gold score.  One thread per batch chain.
// ---------------------------------------------------------------------------
__device__ __forceinline__ float lse4(float x0, float x1, float x2, float x3) {
    float mx = fmaxf(fmaxf(x0, x1), fmaxf(x2, x3));
    return mx + __logf(__expf(x0 - mx) + __expf(x1 - mx) +
                       __expf(x2 - mx) + __expf(x3 - mx));
}

__global__ __launch_bounds__(64) void crf_kernel(
        const int*   __restrict__ words,
        const int*   __restrict__ target,
        const float* __restrict__ trans_m,
        const float* __restrict__ start_s,
        const float* __restrict__ end_s,
        const float* __restrict__ logits,
        float*       __restrict__ out) {
    __shared__ float sT[TT * TT];
    __shared__ float sS[TT];
    __shared__ float sE[TT];
    const int tid = threadIdx.x;
    if (tid < TT * TT) sT[tid] = trans_m[tid];
    if (tid < TT) { sS[tid] = start_s[tid]; sE[tid] = end_s[tid]; }
    __syncthreads();

    const int b = tid;                         // 64 chains
    const float* lg = logits + (long)b * SS * TT;
    const int*   w  = words  + (long)b * SS;
    const int*   tg = target + (long)b * SS;

    float a0 = lg[0] + sS[0];
    float a1 = lg[1] + sS[1];
    float a2 = lg[2] + sS[2];
    float a3 = lg[3] + sS[3];

    const int  t0 = tg[0];
    const bool m0 = (w[0] != 0);
    int   cnt   = m0 ? 1 : 0;
    float emit  = m0 ? lg[t0] : 0.0f;
    float tracc = 0.0f;
    int   tprev = t0;

    for (int s = 1; s < SS; ++s) {
        const float l0 = lg[s * 4 + 0];
        const float l1 = lg[s * 4 + 1];
        const float l2 = lg[s * 4 + 2];
        const float l3 = lg[s * 4 + 3];
        const bool  ms = (w[s] != 0);

        float n0 = lse4(a0 + sT[0], a1 + sT[4], a2 + sT[8],  a3 + sT[12]) + l0;
        float n1 = lse4(a0 + sT[1], a1 + sT[5], a2 + sT[9],  a3 + sT[13]) + l1;
        float n2 = lse4(a0 + sT[2], a1 + sT[6], a2 + sT[10], a3 + sT[14]) + l2;
        float n3 = lse4(a0 + sT[3], a1 + sT[7], a2 + sT[11], a3 + sT[15]) + l3;
        a0 = ms ? n0 : a0;
        a1 = ms ? n1 : a1;
        a2 = ms ? n2 : a2;
        a3 = ms ? n3 : a3;

        const int tc = tg[s];
        emit  += ms ? lg[s * 4 + tc]     : 0.0f;
        tracc += ms ? sT[tprev * 4 + tc] : 0.0f;
        tprev  = tc;
        cnt   += ms ? 1 : 0;
    }

    const float normalizer = lse4(a0 + sE[0], a1 + sE[1], a2 + sE[2], a3 + sE[3]);
    const int   last  = (cnt - 1) > 0 ? (cnt - 1) : 0;
    const int   lastt = tg[last];
    const float gold  = emit + tracc + sS[t0] + sE[lastt];
    out[b] = normalizer - gold;
}

// ---------------------------------------------------------------------------
// Launch wrapper
// ---------------------------------------------------------------------------
extern "C" void kernel_launch(void* const* d_in, const int* in_sizes, int n_in,
                              void* d_out, int out_size, void* d_ws, size_t ws_size,
                              hipStream_t stream) {
    const int*   words    = (const int*)  d_in[0];
    const int*   target   = (const int*)  d_in[1];
    const int*   corpus   = (const int*)  d_in[2];
    const float* embed    = (const float*)d_in[3];
    const float* shared_W = (const float*)d_in[4];
    const float* shared_b = (const float*)d_in[5];
    const float* domain_A = (const float*)d_in[6];
    const float* domain_b = (const float*)d_in[7];
    const float* trans_m  = (const float*)d_in[8];
    const float* start_s  = (const float*)d_in[9];
    const float* end_s    = (const float*)d_in[10];
    float* out = (float*)d_out;

    float* ws     = (float*)d_ws;
    float* WcT    = ws;                       // NC*TT*DD = 30720 floats
    float* bc     = ws + NC * TT * DD;        // NC*TT    = 40 floats
    float* logits = ws + 32768;               // BB*SS*TT = 131072 floats

    prep_kernel<<<(NC * TT * DD + 255) / 256, 256, 0, stream>>>(
        shared_W, shared_b, domain_A, domain_b, WcT, bc);

    proj_kernel<<<(BB * SS) / 128, 256, 0, stream>>>(
        words, corpus, embed, WcT, bc, logits);

    crf_kernel<<<1, 64, 0, stream>>>(
        words, target, trans_m, start_s, end_s, logits, out);
}